// Knowledge_aware_Fusion_Layer_48782238548034
// MI455X (gfx1250) — compile-verified
//
#include <hip/hip_runtime.h>
#include <math.h>

// ---------------------------------------------------------------------------
// Knowledge-aware Fusion Layer for MI455X (gfx1250, wave32, WMMA)
// B=8, H=1024, FF=4096, L=12 per stack, 2 stacks.
// GEMMs: V_WMMA_F32_16X16X4_F32, M padded 8->16, 32-col block tiles,
// K-split partials + epilogue for concurrency. Weights streamed with
// nontemporal loads (768 MB total, ~33us HBM floor at 23.3 TB/s).
// ---------------------------------------------------------------------------

constexpr int   Hn   = 1024;
constexpr int   FFn  = 4096;
constexpr int   Bn   = 8;
constexpr int   Ln   = 12;
constexpr float EPSn = 1e-5f;

typedef __attribute__((ext_vector_type(2))) float v2f;
typedef __attribute__((ext_vector_type(8))) float v8f;

__device__ __forceinline__ float gelu_exact(float x) {
  return 0.5f * x * (1.0f + erff(x * 0.70710678118654752440f));
}

// ---------------------------------------------------------------------------
// part[ky][8,N] += A[8,K-slice] @ W[K-slice, N]  (raw partial sums, no bias)
// Grid: (N/32, KSPLIT). Block = 8 waves; each wave owns a 32-col tile slice
// of K/(8*KSPLIT) (multiple of 16). Two 16x16 WMMA accumulators per wave so
// every W row access consumes a full 128B cacheline -> safe NT streaming.
// Wave32 f32 WMMA layouts (CDNA5 ISA 7.12.2):
//   A 16x4: lane(0..15)=row M, VGPR0 K={0|2} by lane-half, VGPR1 K={1|3}
//   B 4x16: lane(0..15)=col N, same K striping
//   C 16x16: VGPR i -> M=i (lanes 0-15), M=8+i (lanes 16-31)
// ---------------------------------------------------------------------------
template<int K, int N, int KSPLIT>
__global__ __launch_bounds__(256, 2)
void gemm8_wmma_kernel(const float* __restrict__ A, const float* __restrict__ W,
                       float* __restrict__ part)
{
  constexpr int KB = K / KSPLIT;        // K range per y-block
  constexpr int KC = KB / 8;            // K range per wave
  static_assert(KC % 16 == 0, "wave K chunk must be a multiple of 16");

  __shared__ float red[2 * 8 * 8 * 32]; // [tile][c-reg][wave][lane]
  const int tid   = threadIdx.x;
  const int wave  = tid >> 5;
  const int lane  = tid & 31;
  const int hl    = lane >> 4;          // lane half selects K pair
  const int col   = lane & 15;
  const int n0    = blockIdx.x << 5;    // 32-column tile
  const int kbase = blockIdx.y * KB + wave * KC;

  const bool   rvalid = col < Bn;       // rows 8..15 of the 16x4 A are zero
  const float* Ap = A + (size_t)(col & 7) * K + kbase + (hl << 1);
  const float* Wp = W + (size_t)(kbase + (hl << 1)) * N + n0 + col;
  const float* Pf = Wp + (size_t)16 * N;

  v8f c0 = {0.f,0.f,0.f,0.f,0.f,0.f,0.f,0.f};
  v8f c1 = {0.f,0.f,0.f,0.f,0.f,0.f,0.f,0.f};

  for (int k = 0; k < KC; k += 16) {
    __builtin_prefetch(Pf, 0, 1);       // global_prefetch_b8, 16 W-rows ahead
    Pf += (size_t)16 * N;
    #pragma unroll
    for (int u = 0; u < 4; ++u) {
      v2f a = *(const v2f*)(Ap);        // global_load_b64, always in-bounds
      Ap += 4;
      a.x = rvalid ? a.x : 0.f;         // v_cndmask, no EXEC branching
      a.y = rvalid ? a.y : 0.f;
      v2f b0, b1;
      b0.x = __builtin_nontemporal_load(Wp);
      b0.y = __builtin_nontemporal_load(Wp + N);
      b1.x = __builtin_nontemporal_load(Wp + 16);
      b1.y = __builtin_nontemporal_load(Wp + N + 16);
      Wp += (size_t)4 * N;
      c0 = __builtin_amdgcn_wmma_f32_16x16x4_f32(false, a, false, b0,
                                                 (short)0, c0, false, false);
      c1 = __builtin_amdgcn_wmma_f32_16x16x4_f32(false, a, false, b1,
                                                 (short)0, c1, false, false);
    }
  }

  #pragma unroll
  for (int i = 0; i < 8; ++i) {
    red[((0 * 8 + i) * 8 + wave) * 32 + lane] = c0[i];
    red[((1 * 8 + i) * 8 + wave) * 32 + lane] = c1[i];
  }
  __syncthreads();

  if (wave == 0 && lane < 16) {         // lanes 0..15 carry rows M=0..7
    float* out = part + (size_t)blockIdx.y * (Bn * N);
    #pragma unroll
    for (int t = 0; t < 2; ++t) {
      const int n = n0 + t * 16 + lane;
      #pragma unroll
      for (int i = 0; i < 8; ++i) {
        float sv = 0.f;
        #pragma unroll
        for (int w = 0; w < 8; ++w) sv += red[((t * 8 + i) * 8 + w) * 32 + lane];
        out[(size_t)i * N + n] = sv;
      }
    }
  }
}

// ---------------------------------------------------------------------------
// C[8,N] = act( sum_ks part[ks][8,N] + bias ) (+ resid). N power of two.
// ---------------------------------------------------------------------------
__global__ __launch_bounds__(256)
void gemm_epilogue_kernel(const float* __restrict__ part, const float* __restrict__ bias,
                          const float* __restrict__ resid, float* __restrict__ C,
                          int N, int ks, int act)
{
  const int idx = blockIdx.x * 256 + threadIdx.x;   // over Bn*N
  float s = 0.f;
  for (int p = 0; p < ks; ++p) s += part[(size_t)p * Bn * N + idx];
  float val = s + bias[idx & (N - 1)];
  if (act)   val = gelu_exact(val);
  if (resid) val += resid[idx];
  C[idx] = val;
}

// ---------------------------------------------------------------------------
// Per batch row b: v = LN(kv)*g1+be1 ; kp = kv*wk+bk ; vp = v*wv+bv ;
// qp = query*wq+bq ; kstats[b] = {max kp, min kp} (O(1) softmax max).
// ---------------------------------------------------------------------------
__global__ __launch_bounds__(256)
void ln1_proj_kernel(const float* __restrict__ kv, const float* __restrict__ query,
                     const float* __restrict__ inw, const float* __restrict__ inb,
                     const float* __restrict__ g1, const float* __restrict__ be1,
                     float* __restrict__ v, float* __restrict__ kpvp,  // float2 interleaved
                     float* __restrict__ qp, float* __restrict__ kstats)
{
  __shared__ float sa[256], sb[256];
  const int b = blockIdx.x, tid = threadIdx.x;
  const float* x = kv + (size_t)b * Hn;

  float sum = 0.f, sq = 0.f;
  for (int j = tid; j < Hn; j += 256) { float t = x[j]; sum += t; sq += t * t; }
  sa[tid] = sum; sb[tid] = sq; __syncthreads();
  for (int s = 128; s > 0; s >>= 1) {
    if (tid < s) { sa[tid] += sa[tid + s]; sb[tid] += sb[tid + s]; }
    __syncthreads();
  }
  const float mean = sa[0] * (1.f / Hn);
  const float var  = sb[0] * (1.f / Hn) - mean * mean;
  const float rstd = rsqrtf(var + EPSn);
  __syncthreads();

  const float wq = inw[0], wk = inw[1], wv = inw[2];
  const float bq = inb[0], bk = inb[1], bv = inb[2];
  float kmax = -3.4e38f, kmin = 3.4e38f;
  float2* kvp2 = (float2*)kpvp + (size_t)b * Hn;
  for (int j = tid; j < Hn; j += 256) {
    float t  = x[j];
    float lv = (t - mean) * rstd * g1[j] + be1[j];
    v[(size_t)b * Hn + j] = lv;
    float kj = t * wk + bk;
    kvp2[j] = make_float2(kj, lv * wv + bv);
    qp[(size_t)b * Hn + j] = query[(size_t)b * Hn + j] * wq + bq;
    kmax = fmaxf(kmax, kj); kmin = fminf(kmin, kj);
  }
  sa[tid] = kmax; sb[tid] = kmin; __syncthreads();
  for (int s = 128; s > 0; s >>= 1) {
    if (tid < s) { sa[tid] = fmaxf(sa[tid], sa[tid + s]); sb[tid] = fminf(sb[tid], sb[tid + s]); }
    __syncthreads();
  }
  if (tid == 0) { kstats[2 * b] = sa[0]; kstats[2 * b + 1] = sb[0]; }
}

// ---------------------------------------------------------------------------
// y[b,q] = (sum_k softmax_k(qp[b,q]*kp[b,k]) * vp[b,k]) * ow + ob + v[b,q]
// Wave-per-query, lanes split k; interleaved kp/vp staged in LDS (b64 loads);
// v_exp_f32 hot loop. Grid: (H/8, B), 256 threads (8 waves).
// ---------------------------------------------------------------------------
__global__ __launch_bounds__(256)
void attn_kernel(const float* __restrict__ kpvp, const float* __restrict__ qp,
                 const float* __restrict__ v, const float* __restrict__ kstats,
                 const float* __restrict__ ow, const float* __restrict__ ob,
                 float* __restrict__ y)
{
  __shared__ float2 skv[Hn];
  const int b    = blockIdx.y;
  const int tid  = threadIdx.x;
  const int wave = tid >> 5;
  const int lane = tid & 31;
  const float2* g = (const float2*)kpvp + (size_t)b * Hn;
  for (int j = tid; j < Hn; j += 256) skv[j] = g[j];
  __syncthreads();

  const int   q  = (blockIdx.x << 3) + wave;
  const float s  = qp[(size_t)b * Hn + q];
  const float nm = -((s >= 0.f) ? s * kstats[2 * b] : s * kstats[2 * b + 1]);

  float den = 0.f, acc = 0.f;
  #pragma unroll 4
  for (int k = lane; k < Hn; k += 32) {
    float2 t = skv[k];
    float  e = __expf(fmaf(s, t.x, nm));
    den += e;
    acc  = fmaf(e, t.y, acc);
  }
  for (int off = 16; off > 0; off >>= 1) {
    den += __shfl_xor(den, off);
    acc += __shfl_xor(acc, off);
  }
  if (lane == 0)
    y[(size_t)b * Hn + q] = (acc / den) * ow[0] + ob[0] + v[(size_t)b * Hn + q];
}

// ---------------------------------------------------------------------------
// out[b,:] = LN(x[b,:])*g + be   (row length n), one block per row.
// ---------------------------------------------------------------------------
__global__ __launch_bounds__(256)
void ln_row_kernel(const float* __restrict__ x, const float* __restrict__ g,
                   const float* __restrict__ be, float* __restrict__ out, int n)
{
  __shared__ float sa[256], sb[256];
  const int b = blockIdx.x, tid = threadIdx.x;
  const float* xr = x + (size_t)b * n;
  float sum = 0.f, sq = 0.f;
  for (int j = tid; j < n; j += 256) { float t = xr[j]; sum += t; sq += t * t; }
  sa[tid] = sum; sb[tid] = sq; __syncthreads();
  for (int s = 128; s > 0; s >>= 1) {
    if (tid < s) { sa[tid] += sa[tid + s]; sb[tid] += sb[tid + s]; }
    __syncthreads();
  }
  const float mean = sa[0] / n;
  const float rstd = rsqrtf(sb[0] / n - mean * mean + EPSn);
  for (int j = tid; j < n; j += 256)
    out[(size_t)b * n + j] = (xr[j] - mean) * rstd * g[j] + be[j];
}

// ---------------------------------------------------------------------------
// out[b,:] = LN(concat(s1[b,:], s2[b,:])) * g + be  over 2H
// ---------------------------------------------------------------------------
__global__ __launch_bounds__(256)
void final_ln_kernel(const float* __restrict__ s1, const float* __restrict__ s2,
                     const float* __restrict__ g, const float* __restrict__ be,
                     float* __restrict__ out)
{
  const int n = 2 * Hn;
  __shared__ float sa[256], sb[256];
  const int b = blockIdx.x, tid = threadIdx.x;
  float sum = 0.f, sq = 0.f;
  for (int j = tid; j < n; j += 256) {
    float t = (j < Hn) ? s1[(size_t)b * Hn + j] : s2[(size_t)b * Hn + j - Hn];
    sum += t; sq += t * t;
  }
  sa[tid] = sum; sb[tid] = sq; __syncthreads();
  for (int s = 128; s > 0; s >>= 1) {
    if (tid < s) { sa[tid] += sa[tid + s]; sb[tid] += sb[tid + s]; }
    __syncthreads();
  }
  const float mean = sa[0] / n;
  const float rstd = rsqrtf(sb[0] / n - mean * mean + EPSn);
  for (int j = tid; j < n; j += 256) {
    float t = (j < Hn) ? s1[(size_t)b * Hn + j] : s2[(size_t)b * Hn + j - Hn];
    out[(size_t)b * n + j] = (t - mean) * rstd * g[j] + be[j];
  }
}

// ---------------------------------------------------------------------------
extern "C" void kernel_launch(void* const* d_in, const int* in_sizes, int n_in,
                              void* d_out, int out_size, void* d_ws, size_t ws_size,
                              hipStream_t stream)
{
  (void)in_sizes; (void)n_in; (void)out_size; (void)ws_size;

  const float* ct  = (const float*)d_in[0];
  const float* cl  = (const float*)d_in[1];
  const float* Wct = (const float*)d_in[2];
  const float* bct = (const float*)d_in[3];
  const float* Wcl = (const float*)d_in[4];
  const float* bcl = (const float*)d_in[5];
  const float* ng  = (const float*)d_in[6];
  const float* nb  = (const float*)d_in[7];

  float* ws   = (float*)d_ws;
  float* q1   = ws;                 // cle (query of att1)
  float* q2   = q1   + Bn * Hn;     // cte (query of att2)
  float* s1   = q2   + Bn * Hn;     // att1 carried state (starts = cte)
  float* s2   = s1   + Bn * Hn;     // att2 carried state (starts = cle)
  float* vb   = s2   + Bn * Hn;     // LN1(value)
  float* kvpb = vb   + Bn * Hn;     // interleaved {kp,vp}  [B, H, 2]
  float* qpb  = kvpb + 2 * Bn * Hn;
  float* yb   = qpb  + Bn * Hn;     // attn-out + v (pre-LN2)
  float* xb   = yb   + Bn * Hn;     // LN2 output / MLP residual
  float* hb   = xb   + Bn * Hn;     // MLP hidden [8, FF]
  float* kst  = hb   + Bn * FFn;    // per-batch kp max/min [2*B]
  float* pbuf = kst  + 2 * Bn;      // GEMM K-split partials (<= 2*8*FF floats)

  // Embeddings: cte = ct@W_ct+b_ct (-> q2 and s1), cle = clinical@W_cl+b_cl
  gemm8_wmma_kernel<2048, Hn, 2><<<dim3(Hn / 32, 2), 256, 0, stream>>>(ct, Wct, pbuf);
  gemm_epilogue_kernel<<<Bn * Hn / 256, 256, 0, stream>>>(pbuf, bct, nullptr, q2, Hn, 2, 0);
  gemm_epilogue_kernel<<<Bn * Hn / 256, 256, 0, stream>>>(pbuf, bct, nullptr, s1, Hn, 2, 0);

  gemm8_wmma_kernel<128, Hn, 1><<<dim3(Hn / 32, 1), 256, 0, stream>>>(cl, Wcl, pbuf);
  gemm_epilogue_kernel<<<Bn * Hn / 256, 256, 0, stream>>>(pbuf, bcl, nullptr, q1, Hn, 1, 0);
  gemm_epilogue_kernel<<<Bn * Hn / 256, 256, 0, stream>>>(pbuf, bcl, nullptr, s2, Hn, 1, 0);

  for (int st = 0; st < 2; ++st) {
    const int base = 8 + st * 12;
    const float* inw = (const float*)d_in[base + 0];
    const float* inb = (const float*)d_in[base + 1];
    const float* ow  = (const float*)d_in[base + 2];
    const float* ob  = (const float*)d_in[base + 3];
    const float* g1  = (const float*)d_in[base + 4];
    const float* be1 = (const float*)d_in[base + 5];
    const float* g2  = (const float*)d_in[base + 6];
    const float* be2 = (const float*)d_in[base + 7];
    const float* W1  = (const float*)d_in[base + 8];
    const float* b1  = (const float*)d_in[base + 9];
    const float* W2  = (const float*)d_in[base + 10];
    const float* b2  = (const float*)d_in[base + 11];

    float*       state = (st == 0) ? s1 : s2;
    const float* qry   = (st == 0) ? q1 : q2;

    for (int l = 0; l < Ln; ++l) {
      ln1_proj_kernel<<<Bn, 256, 0, stream>>>(
          state, qry, inw + 3 * l, inb + 3 * l,
          g1 + (size_t)l * Hn, be1 + (size_t)l * Hn,
          vb, kvpb, qpb, kst);

      attn_kernel<<<dim3(Hn / 8, Bn), 256, 0, stream>>>(
          kvpb, qpb, vb, kst, ow + l, ob + l, yb);

      ln_row_kernel<<<Bn, 256, 0, stream>>>(
          yb, g2 + (size_t)l * Hn, be2 + (size_t)l * Hn, xb, Hn);

      // MLP1: hidden = gelu(x @ W1 + b1)   [K=1024 -> FF=4096], K-split 2
      gemm8_wmma_kernel<Hn, FFn, 2><<<dim3(FFn / 32, 2), 256, 0, stream>>>(
          xb, W1 + (size_t)l * Hn * FFn, pbuf);
      gemm_epilogue_kernel<<<Bn * FFn / 256, 256, 0, stream>>>(
          pbuf, b1 + (size_t)l * FFn, nullptr, hb, FFn, 2, 1);

      // MLP2: state = x + hidden @ W2 + b2 [K=4096 -> H=1024], K-split 4
      gemm8_wmma_kernel<FFn, Hn, 4><<<dim3(Hn / 32, 4), 256, 0, stream>>>(
          hb, W2 + (size_t)l * FFn * Hn, pbuf);
      gemm_epilogue_kernel<<<Bn * Hn / 256, 256, 0, stream>>>(
          pbuf, b2 + (size_t)l * Hn, xb, state, Hn, 4, 0);
    }
  }

  final_ln_kernel<<<Bn, 256, 0, stream>>>(s1, s2, ng, nb, (float*)d_out);
}